// SignatureCombine_82222853915370
// MI455X (gfx1250) — compile-verified
//
#include <hip/hip_runtime.h>

typedef __attribute__((ext_vector_type(2))) float v2f;
typedef __attribute__((ext_vector_type(8))) float v8f;

#define SIG_LEN   4680   // 8 + 64 + 512 + 4096
#define L2_OFF    8
#define L3_OFF    72
#define L4_OFF    584
#define STAGE_LEN 584    // levels 1..3 are the first 584 floats
#define WPB       4      // waves (batch elements) per block

// One wave32 per batch element. Levels 1-3 of both signatures staged in LDS.
// Level-4 a2(x)b2 rank-1 term computed with V_WMMA_F32_16X16X4_F32 (exact f32:
// A slots hold a/4 in all K, B rows hold b in all K -> D = a*b + C exactly).
__global__ __launch_bounds__(128) void sig_combine_kernel(
    const float* __restrict__ s1, const float* __restrict__ s2,
    float* __restrict__ out)
{
    __shared__ float sA[WPB][STAGE_LEN];
    __shared__ float sB[WPB][STAGE_LEN];

    const int lane = threadIdx.x & 31;
    const int w    = threadIdx.x >> 5;
    const long e   = (long)blockIdx.x * WPB + w;
    const long base = e * SIG_LEN;

    const float* __restrict__ p1 = s1 + base;
    const float* __restrict__ p2 = s2 + base;
    float* __restrict__ po = out + base;

    // Prefetch the level-4 stream (16 KB per signature) while we stage L1-3.
#pragma unroll
    for (int it = 0; it < 4; ++it) {
        __builtin_prefetch(p1 + L4_OFF + (it * 32 + lane) * 32, 0, 3);
        __builtin_prefetch(p2 + L4_OFF + (it * 32 + lane) * 32, 0, 3);
    }

    // Stage levels 1-3 (584 floats each signature) into LDS.
    for (int i = lane; i < STAGE_LEN; i += 32) {
        sA[w][i] = p1[i];
        sB[w][i] = p2[i];
    }
    __syncthreads();

    const float* __restrict__ A  = sA[w];
    const float* __restrict__ Bv = sB[w];

    // ---- Level 1 (8): out = a1 + b1 ----
    if (lane < 8) po[lane] = A[lane] + Bv[lane];

    // ---- Level 2 (64): + a1[i>>3]*b1[i&7] ----
#pragma unroll
    for (int it = 0; it < 2; ++it) {
        int i = it * 32 + lane;
        po[L2_OFF + i] = A[L2_OFF + i] + Bv[L2_OFF + i] + A[i >> 3] * Bv[i & 7];
    }

    // ---- Level 3 (512): + a1[i>>6]*b2[i&63] + a2[i>>3]*b1[i&7] ----
#pragma unroll 4
    for (int it = 0; it < 16; ++it) {
        int i = it * 32 + lane;
        float v = A[L3_OFF + i] + Bv[L3_OFF + i];
        v = fmaf(A[i >> 6], Bv[L2_OFF + (i & 63)], v);
        v = fmaf(A[L2_OFF + (i >> 3)], Bv[i & 7], v);
        po[L3_OFF + i] = v;
    }

    // ---- Level 4 (4096) as a 64x64 matrix in 16x16 WMMA tiles ----
    // out4[r*64+c] = s1_4 + s2_4 + a1[r>>3]*b3[(r&7)*64+c]
    //                            + a3[r*8+(c>>3)]*b1[c&7]  (VALU, into C)
    //                            + a2[r]*b2[c]             (WMMA)
    const int ln   = lane & 15;
    const int half = lane >> 4;              // C/D layout: rows j / j+8
    const float b1v = Bv[ln & 7];            // c&7 == ln&7, tile-invariant

#pragma unroll
    for (int tm = 0; tm < 4; ++tm) {
        const float a1v = A[tm * 2 + half];  // r>>3 == tm*2 + half
        const float av  = A[L2_OFF + tm * 16 + ln] * 0.25f;
        const v2f afrag = {av, av};          // a2[m]/4 in every K slot
#pragma unroll
        for (int tn = 0; tn < 4; ++tn) {
            const int c = tn * 16 + ln;
            const float bvv = Bv[L2_OFF + tn * 16 + ln];
            const v2f bfrag = {bvv, bvv};    // b2[n] in every K row

            v8f cacc;
#pragma unroll
            for (int j = 0; j < 8; ++j) {
                const int r = tm * 16 + half * 8 + j;
                const int i4 = r * 64 + c;
                float v = p1[L4_OFF + i4] + p2[L4_OFF + i4];
                v = fmaf(a1v, Bv[L3_OFF + j * 64 + c], v);                      // a1 (x) b3
                v = fmaf(A[L3_OFF + tm * 128 + half * 64 + j * 8 + (c >> 3)],   // a3 (x) b1
                         b1v, v);
                cacc[j] = v;
            }

            // D = (a2/4) * b2 summed over 4 identical K slots + C = a2(x)b2 + C
            v8f d = __builtin_amdgcn_wmma_f32_16x16x4_f32(
                false, afrag, false, bfrag, (short)0, cacc, false, false);

#pragma unroll
            for (int j = 0; j < 8; ++j) {
                const int r = tm * 16 + half * 8 + j;
                po[L4_OFF + r * 64 + c] = d[j];
            }
        }
    }
}

extern "C" void kernel_launch(void* const* d_in, const int* in_sizes, int n_in,
                              void* d_out, int out_size, void* d_ws, size_t ws_size,
                              hipStream_t stream) {
    const float* s1 = (const float*)d_in[0];
    const float* s2 = (const float*)d_in[1];
    float* out = (float*)d_out;
    const int B = in_sizes[0] / SIG_LEN;       // 16384 for the harness
    const int blocks = B / WPB;                // B divisible by 4
    sig_combine_kernel<<<blocks, 128, 0, stream>>>(s1, s2, out);
}